// ALSHConv_40896678592534
// MI455X (gfx1250) — compile-verified
//
#include <hip/hip_runtime.h>
#include <math.h>

typedef __attribute__((ext_vector_type(2))) float v2f;
typedef __attribute__((ext_vector_type(8))) float v8f;

#define O_CH  256
#define C_CH  64
#define TAPS  9
#define ALEN  585            // D*K*K = 65*9
#define FLAT  576            // C*K*K = 64*9
#define MPOW  9
#define TBL   8192
#define HW    256
#define TILE  14
#define GRID  16             // TILE + 2 halo
#define NTILE 19             // ceil(256/14)

#define OUT_ACTIVE 0
#define OUT_COUNT  (2 * O_CH * FLAT)     // 294912
#define OUT_INDEX  (OUT_COUNT + TBL)     // 303104

// ---------------------------------------------------------------------------
// Zero the histogram region of d_out (harness poisons buffers with 0xAA).
// ---------------------------------------------------------------------------
__global__ void alsh_zero(float* __restrict__ count) {
  int i = blockIdx.x * blockDim.x + threadIdx.x;
  if (i < TBL) count[i] = 0.0f;
}

// ---------------------------------------------------------------------------
// Vote kernel: per-block 16x16 halo grid of S[pixel][tap] via WMMA f32
// 16x16x4.  B (the hash vector arranged [4ch x 9taps]) is hoisted into
// registers once; per grid row all 32 A loads issue as one clause, then the
// 16 WMMAs run back-to-back on two independent accumulators.
// ---------------------------------------------------------------------------
__global__ __launch_bounds__(256) void alsh_vote(
    const float* __restrict__ inp,   // [N, 64, 256, 256]
    const float* __restrict__ av,    // [585]
    const float* __restrict__ bv,    // [1]
    float* __restrict__ count)       // [8192]
{
  __shared__ float sa[ALEN];
  __shared__ float S[GRID * GRID * TAPS];   // [gy][m][tap]
  __shared__ int   hist[TBL];

  const int t    = (int)threadIdx.x;
  const int wv   = t >> 5;         // wave id 0..7
  const int lane = t & 31;
  const int half = lane >> 4;      // 0: lanes 0-15, 1: lanes 16-31
  const int nc   = lane & 15;      // A: pixel M index; B/C: tap column N

  for (int i = t; i < TBL; i += 256) hist[i] = 0;
  for (int i = t; i < ALEN; i += 256) sa[i] = av[i];
  __syncthreads();

  const int n     = (int)blockIdx.z;
  const int xbase = (int)blockIdx.x * TILE;
  const int ybase = (int)blockIdx.y * TILE;
  const size_t imgbase = (size_t)n * C_CH * HW * HW;

  // ---- hoist B into registers: lane-invariant across rows & chunks ----
  float Bx[16], By[16];
  if (nc < TAPS) {
#pragma unroll
    for (int ch = 0; ch < 16; ++ch) {
      const int c0 = ch * 4 + half * 2;
      Bx[ch] = sa[(c0 + 0) * TAPS + nc];
      By[ch] = sa[(c0 + 1) * TAPS + nc];
    }
  } else {
#pragma unroll
    for (int ch = 0; ch < 16; ++ch) { Bx[ch] = 0.0f; By[ch] = 0.0f; }
  }
  const float bias = (nc < TAPS) ? 0.5f * sa[FLAT + nc] : 0.0f;

  const int  acol = xbase - 1 + nc;                 // A pixel column (M = nc)
  const bool colv = (acol >= 0) && (acol < HW);

  // Each wave computes two grid rows: gy = wv and wv + 8.
  for (int row = 0; row < 2; ++row) {
    const int  gy = wv + row * 8;
    const int  y  = ybase - 1 + gy;
    const bool rv = (y >= 0) && (y < HW);
    const bool cv = rv && colv;

    // ---- one clause of 32 global loads for this row's A chunks ----
    float Ax[16], Ay[16];
    if (cv) {
      const float* p = inp + imgbase + (size_t)(half * 2) * (HW * HW)
                       + (size_t)y * HW + acol;
#pragma unroll
      for (int ch = 0; ch < 16; ++ch) {
        const size_t co = (size_t)(ch * 4) * (HW * HW);
        Ax[ch] = p[co];
        Ay[ch] = p[co + HW * HW];
      }
      __builtin_prefetch(p + HW, 0, 1);   // hint: next image row
    } else {
#pragma unroll
      for (int ch = 0; ch < 16; ++ch) { Ax[ch] = 0.0f; Ay[ch] = 0.0f; }
    }

    // ---- 16 back-to-back WMMAs, two independent accumulation chains ----
    v8f acc0 = {0.f, 0.f, 0.f, 0.f, 0.f, 0.f, 0.f, 0.f};
    v8f acc1 = {0.f, 0.f, 0.f, 0.f, 0.f, 0.f, 0.f, 0.f};
#pragma unroll
    for (int ch = 0; ch < 16; ch += 2) {
      v2f A, B;
      A.x = Ax[ch];     A.y = Ay[ch];
      B.x = Bx[ch];     B.y = By[ch];
      acc0 = __builtin_amdgcn_wmma_f32_16x16x4_f32(
          false, A, false, B, (short)0, acc0, false, false);
      A.x = Ax[ch + 1]; A.y = Ay[ch + 1];
      B.x = Bx[ch + 1]; B.y = By[ch + 1];
      acc1 = __builtin_amdgcn_wmma_f32_16x16x4_f32(
          false, A, false, B, (short)0, acc1, false, false);
    }

    // ---- scatter C into LDS: lane holds tap column nc, VGPR r = pixel m ----
    if (nc < TAPS) {
#pragma unroll
      for (int r = 0; r < 8; ++r) {
        const int  m  = r + 8 * half;
        const int  pc = xbase - 1 + m;
        const bool pv = rv && (pc >= 0) && (pc < HW);
        S[(gy * GRID + m) * TAPS + nc] = pv ? (acc0[r] + acc1[r] + bias) : 0.0f;
      }
    }
  }
  __syncthreads();

  // ---- stencil combine over the 14x14 interior + vote ----
  const float bb = bv[0];
  if (t < TILE * TILE) {
    const int oy = t / TILE, ox = t % TILE;
    const int y = ybase + oy, x = xbase + ox;
    if (y < HW && x < HW) {
      float d = 0.0f;
#pragma unroll
      for (int ky = 0; ky < 3; ++ky)
#pragma unroll
        for (int kx = 0; kx < 3; ++kx)
          d += S[((oy + ky) * GRID + (ox + kx)) * TAPS + (ky * 3 + kx)];
      const float vote = floorf((d + bb) * 0.25f);          // R = 4.0
      const int   bin  = (int)fabsf(fmodf(vote, (float)TBL));
      atomicAdd(&hist[bin], 1);
    }
  }
  __syncthreads();

  for (int i = t; i < TBL; i += 256) {
    const int h = hist[i];
    if (h) atomicAdd(&count[i], (float)h);
  }
}

// ---------------------------------------------------------------------------
// Finalize: argmax(count) (first-max tie-break), hash all 256 kernels,
// gather the winning bucket in ascending kernel-id order (== reference's
// cumsum insertion order + sort), write active / index.
// ---------------------------------------------------------------------------
__global__ __launch_bounds__(256) void alsh_finalize(
    const float* __restrict__ kern,   // [256, 576]
    const float* __restrict__ av,     // [585]
    const float* __restrict__ bv,     // [1]
    const float* __restrict__ count,  // [8192] (final)
    float* __restrict__ out)
{
  __shared__ float sa[ALEN];
  __shared__ float rmax[256];
  __shared__ int   ridx[256];
  __shared__ int   kidx[O_CH];
  __shared__ int   rows[2 * O_CH];

  const int t = (int)threadIdx.x;
  for (int i = t; i < ALEN; i += 256) sa[i] = av[i];

  // ---- argmax (strided scan keeps earliest index within subset) ----
  float best = -1.0f; int bi = TBL;
  for (int i = t; i < TBL; i += 256) {
    const float c = count[i];
    if (c > best) { best = c; bi = i; }
  }
  rmax[t] = best; ridx[t] = bi;
  __syncthreads();
  for (int s = 128; s > 0; s >>= 1) {
    if (t < s) {
      if (rmax[t + s] > rmax[t] ||
          (rmax[t + s] == rmax[t] && ridx[t + s] < ridx[t])) {
        rmax[t] = rmax[t + s]; ridx[t] = ridx[t + s];
      }
    }
    __syncthreads();
  }
  const int   chosen = ridx[0];
  const float bb     = bv[0];

  // ---- hash kernel t (t == kernel id, 256 threads) ----
  {
    const float* kf = kern + (size_t)t * FLAT;
    float dot = 0.0f, n2 = 0.0f;
    for (int i = 0; i < FLAT; ++i) {
      const float v = kf[i];
      dot += v * sa[i];
      n2  += v * v;
    }
    float p = n2;
    for (int m = 0; m < MPOW; ++m) { dot += p * sa[FLAT + m]; p *= n2; }
    const float hk = floorf((dot + bb) * 0.25f);
    kidx[t] = (int)fabsf(fmodf(hk, (float)TBL));
  }
  rows[t] = -1;
  rows[t + O_CH] = -1;
  __syncthreads();

  if (kidx[t] == chosen) {
    int rank = 0;
    for (int o = 0; o < t; ++o) rank += (kidx[o] == chosen) ? 1 : 0;
    rows[rank] = t;                 // ascending kernel id == sorted rows
  }
  __syncthreads();

  // ---- write active [512, 576] ----
  for (int e = t; e < 2 * O_CH * FLAT; e += 256) {
    const int s = e / FLAT;
    const int i = e - s * FLAT;
    const int r = rows[s];
    out[OUT_ACTIVE + e] = (r >= 0) ? kern[(size_t)r * FLAT + i] : 0.0f;
  }
  if (t == 0) out[OUT_INDEX] = (float)chosen;
}

// ---------------------------------------------------------------------------
extern "C" void kernel_launch(void* const* d_in, const int* in_sizes, int n_in,
                              void* d_out, int out_size, void* d_ws, size_t ws_size,
                              hipStream_t stream) {
  const float* inp  = (const float*)d_in[0];   // [N,64,256,256]
  const float* kern = (const float*)d_in[1];   // [256,64,3,3]
  const float* av   = (const float*)d_in[2];   // [585]
  const float* bv   = (const float*)d_in[3];   // [1]
  float* out = (float*)d_out;

  const int N = in_sizes[0] / (C_CH * HW * HW);
  float* count = out + OUT_COUNT;

  alsh_zero<<<(TBL + 255) / 256, 256, 0, stream>>>(count);

  dim3 grid(NTILE, NTILE, N);
  alsh_vote<<<grid, 256, 0, stream>>>(inp, av, bv, count);

  alsh_finalize<<<1, 256, 0, stream>>>(kern, av, bv, count, out);
}